// CrossAttention_76441827934401
// MI455X (gfx1250) — compile-verified
//
#include <hip/hip_runtime.h>
#include <hip/hip_bf16.h>
#include <math.h>

// ---------------------------------------------------------------------------
// Lorentz-equivariant cross attention for MI455X (gfx1250, wave32).
// Full fp32 pipeline; attention GEMMs via V_WMMA_F32_16X16X4_F32.
// Softmax row reductions via fused DPP16 butterflies (1 VALU op per step,
// inline asm since the builtins can't fuse DPP into max/add).
// Q is pre-scaled by 0.25*log2(e) so softmax runs in base-2 (bare v_exp_f32).
// Workspace (floats): Qc | Kc | Vt (transposed) | Hout, 4 x 512K = 8 MB.
// ---------------------------------------------------------------------------

#define B_    2
#define NQ_   2048
#define NKV_  2048
#define H_    8
#define HV_   2
#define HS_   8
#define QV_   16
#define QS_   64
#define DH_   16          // head dim = HV_*4 + HS_
#define EPS_  1e-6f
#define QSCALE_ 0.36067376022224085f   // (1/sqrt(16)) * log2(e)

#if __has_builtin(__builtin_amdgcn_exp2f)
#define EXP2F(x) __builtin_amdgcn_exp2f(x)
#else
#define EXP2F(x) exp2f(x)
#endif

typedef __attribute__((ext_vector_type(2))) float v2f;
typedef __attribute__((ext_vector_type(8))) float v8f;

// ---- fused DPP16 butterfly reductions within each 16-lane half ------------
// One v_max_num_f32_dpp / v_add_f32_dpp per step: dst = op(dpp(x), x).
__device__ __forceinline__ float red_max16(float x) {
    float y;
    asm("v_max_num_f32_dpp %0, %1, %1 quad_perm:[1,0,3,2] row_mask:0xf bank_mask:0xf bound_ctrl:1"
        : "=v"(y) : "v"(x));
    asm("v_max_num_f32_dpp %0, %1, %1 quad_perm:[2,3,0,1] row_mask:0xf bank_mask:0xf bound_ctrl:1"
        : "=v"(x) : "v"(y));
    asm("v_max_num_f32_dpp %0, %1, %1 row_half_mirror row_mask:0xf bank_mask:0xf bound_ctrl:1"
        : "=v"(y) : "v"(x));
    asm("v_max_num_f32_dpp %0, %1, %1 row_mirror row_mask:0xf bank_mask:0xf bound_ctrl:1"
        : "=v"(x) : "v"(y));
    return x;
}
__device__ __forceinline__ float red_sum16(float x) {
    float y;
    asm("v_add_f32_dpp %0, %1, %1 quad_perm:[1,0,3,2] row_mask:0xf bank_mask:0xf bound_ctrl:1"
        : "=v"(y) : "v"(x));
    asm("v_add_f32_dpp %0, %1, %1 quad_perm:[2,3,0,1] row_mask:0xf bank_mask:0xf bound_ctrl:1"
        : "=v"(x) : "v"(y));
    asm("v_add_f32_dpp %0, %1, %1 row_half_mirror row_mask:0xf bank_mask:0xf bound_ctrl:1"
        : "=v"(y) : "v"(x));
    asm("v_add_f32_dpp %0, %1, %1 row_mirror row_mask:0xf bank_mask:0xf bound_ctrl:1"
        : "=v"(x) : "v"(y));
    return x;
}

// ---------------- Kernel 1: Q projection + RMS norm + metric ----------------
// Output pre-scaled by QSCALE_ so attention logits are already base-2.
__global__ __launch_bounds__(256)
void proj_q_kernel(const float* __restrict__ vec, const float* __restrict__ scal,
                   const float* __restrict__ Wv, const float* __restrict__ Ws,
                   const float* __restrict__ bs, float* __restrict__ Qc) {
    int t = blockIdx.x * blockDim.x + threadIdx.x;
    if (t >= B_ * NQ_) return;
    int b = t / NQ_, nq = t % NQ_;

    float vin[QV_][4];
    const float* vp = vec + (size_t)t * QV_ * 4;
    #pragma unroll
    for (int c = 0; c < QV_; ++c)
        #pragma unroll
        for (int d = 0; d < 4; ++d) vin[c][d] = vp[c * 4 + d];

    float sin_[QS_];
    const float* sp = scal + (size_t)t * QS_;
    #pragma unroll
    for (int c = 0; c < QS_; ++c) sin_[c] = sp[c];

    const float metric[4] = {1.f, -1.f, -1.f, -1.f};

    for (int h = 0; h < H_; ++h) {
        float vout[HV_][4];
        #pragma unroll
        for (int hv = 0; hv < HV_; ++hv) {
            int o = hv * H_ + h;                        // flat o = hv*H + h
            #pragma unroll
            for (int d = 0; d < 4; ++d) {
                float a = 0.f;
                #pragma unroll
                for (int c = 0; c < QV_; ++c) a += Wv[o * QV_ + c] * vin[c][d];
                vout[hv][d] = a;
            }
        }
        float sout[HS_];
        #pragma unroll
        for (int hs = 0; hs < HS_; ++hs) {
            int o = hs * H_ + h;                        // flat o = hs*H + h
            float a = bs[o];
            #pragma unroll
            for (int c = 0; c < QS_; ++c) a += Ws[o * QS_ + c] * sin_[c];
            sout[hs] = a;
        }
        float mi = 0.f;
        #pragma unroll
        for (int hv = 0; hv < HV_; ++hv) {
            float inr = vout[hv][0]*vout[hv][0] - vout[hv][1]*vout[hv][1]
                      - vout[hv][2]*vout[hv][2] - vout[hv][3]*vout[hv][3];
            mi += fabsf(inr);
        }
        float vni = rsqrtf(mi / (float)HV_ + EPS_) * QSCALE_;   // fold softmax scale
        float ss = 0.f;
        #pragma unroll
        for (int hs = 0; hs < HS_; ++hs) ss += sout[hs] * sout[hs];
        float sni = rsqrtf(ss / (float)HS_ + EPS_) * QSCALE_;   // fold softmax scale

        float* qp = Qc + (((size_t)(b * H_ + h) * NQ_) + nq) * DH_;
        #pragma unroll
        for (int hv = 0; hv < HV_; ++hv)
            #pragma unroll
            for (int d = 0; d < 4; ++d)
                qp[hv * 4 + d] = vout[hv][d] * metric[d] * vni;  // metric-weighted q
        #pragma unroll
        for (int hs = 0; hs < HS_; ++hs) qp[8 + hs] = sout[hs] * sni;
    }
}

// ---------------- Kernel 2: K/V projection + RMS norm ----------------------
// K stored row-major (key, dh); V stored TRANSPOSED (dh, key) so the P*V
// B-matrix loads in attn are aligned b64 loads of two consecutive keys.
__global__ __launch_bounds__(256)
void proj_kv_kernel(const float* __restrict__ vec, const float* __restrict__ scal,
                    const float* __restrict__ Wv, const float* __restrict__ Ws,
                    const float* __restrict__ bs,
                    float* __restrict__ Kc, float* __restrict__ Vt) {
    int t = blockIdx.x * blockDim.x + threadIdx.x;
    if (t >= B_ * NKV_) return;
    int b = t / NKV_, nk = t % NKV_;

    float vin[QV_][4];
    const float* vp = vec + (size_t)t * QV_ * 4;
    #pragma unroll
    for (int c = 0; c < QV_; ++c)
        #pragma unroll
        for (int d = 0; d < 4; ++d) vin[c][d] = vp[c * 4 + d];

    float sin_[QS_];
    const float* sp = scal + (size_t)t * QS_;
    #pragma unroll
    for (int c = 0; c < QS_; ++c) sin_[c] = sp[c];

    for (int tt = 0; tt < 2; ++tt) {          // 0 = K, 1 = V
        for (int h = 0; h < H_; ++h) {
            float vout[HV_][4];
            #pragma unroll
            for (int hv = 0; hv < HV_; ++hv) {
                int o = tt * (HV_ * H_) + hv * H_ + h;   // Wv_kv: (2*HV*H, KVV)
                #pragma unroll
                for (int d = 0; d < 4; ++d) {
                    float a = 0.f;
                    #pragma unroll
                    for (int c = 0; c < QV_; ++c) a += Wv[o * QV_ + c] * vin[c][d];
                    vout[hv][d] = a;
                }
            }
            float sout[HS_];
            #pragma unroll
            for (int hs = 0; hs < HS_; ++hs) {
                int o = tt * (HS_ * H_) + hs * H_ + h;   // Ws_kv: (2*HS*H, KVS)
                float a = bs[o];
                #pragma unroll
                for (int c = 0; c < QS_; ++c) a += Ws[o * QS_ + c] * sin_[c];
                sout[hs] = a;
            }
            float mi = 0.f;
            #pragma unroll
            for (int hv = 0; hv < HV_; ++hv) {
                float inr = vout[hv][0]*vout[hv][0] - vout[hv][1]*vout[hv][1]
                          - vout[hv][2]*vout[hv][2] - vout[hv][3]*vout[hv][3];
                mi += fabsf(inr);
            }
            float vni = rsqrtf(mi / (float)HV_ + EPS_);
            float ss = 0.f;
            #pragma unroll
            for (int hs = 0; hs < HS_; ++hs) ss += sout[hs] * sout[hs];
            float sni = rsqrtf(ss / (float)HS_ + EPS_);

            float row[DH_];
            #pragma unroll
            for (int hv = 0; hv < HV_; ++hv)
                #pragma unroll
                for (int d = 0; d < 4; ++d) row[hv * 4 + d] = vout[hv][d] * vni;
            #pragma unroll
            for (int hs = 0; hs < HS_; ++hs) row[8 + hs] = sout[hs] * sni;

            if (tt == 0) {                    // K: row-major (key, dh)
                float* p = Kc + (((size_t)(b * H_ + h) * NKV_) + nk) * DH_;
                #pragma unroll
                for (int i = 0; i < DH_; ++i) p[i] = row[i];
            } else {                          // V: transposed (dh, key) — coalesced
                float* p = Vt + ((size_t)(b * H_ + h) * DH_) * NKV_ + nk;
                #pragma unroll
                for (int i = 0; i < DH_; ++i) p[(size_t)i * NKV_] = row[i];
            }
        }
    }
}

// ---------------- Kernel 3: flash attention, fp32 WMMA ---------------------
// One wave per (b, h, 16-query tile). 2048 waves exactly -> grid 256 x 256,
// zero divergence (EXEC all-1s for every WMMA).
__global__ __launch_bounds__(256)
void attn_kernel(const float* __restrict__ Qc, const float* __restrict__ Kc,
                 const float* __restrict__ Vt, float* __restrict__ Hout) {
    __shared__ __align__(16) float ldsP[8][16 * 16];   // per-wave P tile bounce

    const int lane  = threadIdx.x & 31;
    const int wslot = threadIdx.x >> 5;
    const int wid   = blockIdx.x * 8 + wslot;          // 0..2047
    const int qt    = wid & (NQ_ / 16 - 1);            // 0..127
    const int bh    = wid >> 7;                        // 0..15
    const int q0    = qt * 16;

    const int half = lane >> 4;                        // 0: K=0/1 half, 1: K=2/3 half
    const int l15  = lane & 15;
    const int kh   = half * 2;

    const float* Qb  = Qc + ((size_t)bh * NQ_ + q0) * DH_;
    const float* Kb  = Kc + (size_t)bh * NKV_ * DH_;
    const float* Vbt = Vt + (size_t)bh * DH_ * NKV_;   // (dh, key)

    // Q tile in A-layout (16x4 f32 per chunk; lanes0-15:K=0/1, lanes16-31:K=2/3)
    v2f aq[4];
    #pragma unroll
    for (int kk = 0; kk < 4; ++kk)
        aq[kk] = *(const v2f*)(Qb + (size_t)l15 * DH_ + 4 * kk + kh);

    v8f acc = {};
    float rowmax[8], rowsum[8];
    #pragma unroll
    for (int j = 0; j < 8; ++j) { rowmax[j] = -__builtin_inff(); rowsum[j] = 0.f; }

    for (int k0 = 0; k0 < NKV_; k0 += 16) {
        int pk = (k0 + 16 < NKV_) ? (k0 + 16) : 0;
        __builtin_prefetch(Kb + (size_t)(pk + l15) * DH_, 0, 1);
        __builtin_prefetch(Vbt + (size_t)l15 * NKV_ + pk, 0, 1);

        // K tile in B-layout: { K[key][4kk+kh], K[key][4kk+kh+1] }, key = k0+l15
        v2f bk[4];
        #pragma unroll
        for (int kk = 0; kk < 4; ++kk)
            bk[kk] = *(const v2f*)(Kb + (size_t)(k0 + l15) * DH_ + 4 * kk + kh);

        // S = Q * K^T  (16x16 tile, fp32, d=16 contraction = 4 x K=4 WMMA)
        // Q carries 0.25*log2(e): S is already a base-2 logit.
        v8f S = {};
        #pragma unroll
        for (int kk = 0; kk < 4; ++kk)
            S = __builtin_amdgcn_wmma_f32_16x16x4_f32(false, aq[kk], false, bk[kk],
                                                      (short)0, S, false, false);

        // online softmax in base-2; C/D layout: VGPR j -> row (j+8*half), col=l15
        v8f P;
        #pragma unroll
        for (int j = 0; j < 8; ++j) {
            float s    = S[j];
            float tmax = red_max16(s);
            float nm   = fmaxf(rowmax[j], tmax);
            float corr = EXP2F(rowmax[j] - nm);
            rowmax[j]  = nm;
            float p    = EXP2F(s - nm);
            P[j]       = p;
            rowsum[j]  = rowsum[j] * corr + red_sum16(p);
            acc[j]    *= corr;
        }

        // D-layout -> A-layout for P via per-wave LDS tile (DS in-order per wave)
        #pragma unroll
        for (int j = 0; j < 8; ++j)
            ldsP[wslot][(j + half * 8) * 16 + l15] = P[j];
        asm volatile("s_wait_dscnt 0x0" ::: "memory");
        v2f ap[4];
        #pragma unroll
        for (int kk = 0; kk < 4; ++kk)
            ap[kk] = *(const v2f*)&ldsP[wslot][l15 * 16 + 4 * kk + kh];

        // V tile in B-layout from transposed V: two consecutive keys = one b64
        v2f bv[4];
        #pragma unroll
        for (int kk = 0; kk < 4; ++kk)
            bv[kk] = *(const v2f*)(Vbt + (size_t)l15 * NKV_ + k0 + 4 * kk + kh);

        // acc += P * V
        #pragma unroll
        for (int kk = 0; kk < 4; ++kk)
            acc = __builtin_amdgcn_wmma_f32_16x16x4_f32(false, ap[kk], false, bv[kk],
                                                        (short)0, acc, false, false);

        asm volatile("" ::: "memory");   // keep iter i LDS reads before iter i+1 writes
    }

    // finalize: divide by softmax denominator, store head output rows
    float* Hp = Hout + ((size_t)bh * NQ_ + q0) * DH_;
    #pragma unroll
    for (int j = 0; j < 8; ++j) {
        int row = j + half * 8;
        Hp[(size_t)row * DH_ + l15] = acc[j] / rowsum[j];
    }
}

// ---------------- Kernel 4: output projection ------------------------------
__global__ __launch_bounds__(256)
void proj_out_kernel(const float* __restrict__ Hout, const float* __restrict__ Wv,
                     const float* __restrict__ Ws, const float* __restrict__ bs,
                     float* __restrict__ out) {
    int t = blockIdx.x * blockDim.x + threadIdx.x;
    if (t >= B_ * NQ_) return;
    int b = t / NQ_, nq = t % NQ_;

    float hrow[H_][DH_];
    #pragma unroll
    for (int h = 0; h < H_; ++h) {
        const float* p = Hout + (((size_t)(b * H_ + h) * NQ_) + nq) * DH_;
        #pragma unroll
        for (int i = 0; i < DH_; ++i) hrow[h][i] = p[i];
    }

    // out_v: (B,NQ,QV,4), channel c = h*HV + hv, data at hrow[h][hv*4+d]
    float* out_v = out + (size_t)t * QV_ * 4;
    for (int o = 0; o < QV_; ++o) {
        float a[4] = {0.f, 0.f, 0.f, 0.f};
        #pragma unroll
        for (int h = 0; h < H_; ++h)
            #pragma unroll
            for (int hv = 0; hv < HV_; ++hv) {
                float w = Wv[o * QV_ + h * HV_ + hv];
                #pragma unroll
                for (int d = 0; d < 4; ++d) a[d] += w * hrow[h][hv * 4 + d];
            }
        #pragma unroll
        for (int d = 0; d < 4; ++d) out_v[o * 4 + d] = a[d];
    }

    // out_s: (B,NQ,QS), channel c = h*HS + hs, data at hrow[h][8+hs]
    float* out_s = out + (size_t)B_ * NQ_ * QV_ * 4 + (size_t)t * QS_;
    for (int o = 0; o < QS_; ++o) {
        float a = bs[o];
        #pragma unroll
        for (int h = 0; h < H_; ++h)
            #pragma unroll
            for (int hs = 0; hs < HS_; ++hs)
                a += Ws[o * QS_ + h * HS_ + hs] * hrow[h][8 + hs];
        out_s[o] = a;
    }
}

// ---------------------------------------------------------------------------
extern "C" void kernel_launch(void* const* d_in, const int* in_sizes, int n_in,
                              void* d_out, int out_size, void* d_ws, size_t ws_size,
                              hipStream_t stream) {
    const float* vectors    = (const float*)d_in[0];
    const float* scalars    = (const float*)d_in[1];
    const float* vectors_c  = (const float*)d_in[2];
    const float* scalars_c  = (const float*)d_in[3];
    const float* Wv_q  = (const float*)d_in[4];
    const float* Ws_q  = (const float*)d_in[5];
    const float* bs_q  = (const float*)d_in[6];
    const float* Wv_kv = (const float*)d_in[7];
    const float* Ws_kv = (const float*)d_in[8];
    const float* bs_kv = (const float*)d_in[9];
    const float* Wv_o  = (const float*)d_in[10];
    const float* Ws_o  = (const float*)d_in[11];
    const float* bs_o  = (const float*)d_in[12];

    // workspace: 4 head-packed tensors of B*H*N*16 floats = 2 MB each (8 MB total)
    const size_t SEG = (size_t)B_ * H_ * NQ_ * DH_;   // 524288 floats
    float* Qc = (float*)d_ws;
    float* Kc = Qc + SEG;
    float* Vt = Kc + SEG;
    float* Ho = Vt + SEG;

    dim3 blk(256);
    proj_q_kernel<<<(B_ * NQ_ + 255) / 256, blk, 0, stream>>>(
        vectors, scalars, Wv_q, Ws_q, bs_q, Qc);
    proj_kv_kernel<<<(B_ * NKV_ + 255) / 256, blk, 0, stream>>>(
        vectors_c, scalars_c, Wv_kv, Ws_kv, bs_kv, Kc, Vt);
    attn_kernel<<<256, blk, 0, stream>>>(Qc, Kc, Vt, Ho);   // 2048 waves exactly
    proj_out_kernel<<<(B_ * NQ_ + 255) / 256, blk, 0, stream>>>(
        Ho, Wv_o, Ws_o, bs_o, (float*)d_out);
}